// CLUST_GCN_VS_42125039239516
// MI455X (gfx1250) — compile-verified
//
#include <hip/hip_runtime.h>

#define NEG_SLOPE 0.2f
#define LAM 0.2f

typedef __attribute__((ext_vector_type(2))) float v2f;
typedef __attribute__((ext_vector_type(8))) float v8f;

// ---------------- WMMA helper (fp32, 16x16x4) ----------------
__device__ __forceinline__ v8f wmma_f32_k4(v2f a, v2f b, v8f c) {
  // 8 args: (neg_a, A, neg_b, B, c_mod, C, reuse_a, reuse_b)
  return __builtin_amdgcn_wmma_f32_16x16x4_f32(false, a, false, b, (short)0, c,
                                               false, false);
}

// ---------------- float <-> ordered-uint (for atomic float max) ----------------
__device__ __forceinline__ unsigned f2ord(float f) {
  unsigned u = __float_as_uint(f);
  return (u & 0x80000000u) ? ~u : (u | 0x80000000u);
}
__device__ __forceinline__ float ord2f(unsigned u) {
  return (u & 0x80000000u) ? __uint_as_float(u & 0x7FFFFFFFu)
                           : __uint_as_float(~u);
}
#define ORD_NEG_INF 0x007FFFFFu  // f2ord(-inf)

// ---------------- fills ----------------
__global__ void fill_f32_kernel(float* p, long long n, float v) {
  long long i = (long long)blockIdx.x * blockDim.x + threadIdx.x;
  if (i < n) p[i] = v;
}
__global__ void fill_u32_kernel(unsigned* p, long long n, unsigned v) {
  long long i = (long long)blockIdx.x * blockDim.x + threadIdx.x;
  if (i < n) p[i] = v;
}

// ---------------- degree (incl. self loop) ----------------
__global__ void deg_count_kernel(const int* __restrict__ dst, float* deg, int E) {
  int i = blockIdx.x * blockDim.x + threadIdx.x;
  if (i < E) atomicAdd(deg + dst[i], 1.0f);
}
__global__ void recip_kernel(float* deg, int n) {
  int i = blockIdx.x * blockDim.x + threadIdx.x;
  if (i < n) {
    float d = deg[i];
    deg[i] = 1.0f / (d < 1.0f ? 1.0f : d);
  }
}

// ---------------- ClusterGCN aggregation: agg[d] += w * x[s] ----------------
__global__ void cgcn_scatter_kernel(const float* __restrict__ x,
                                    const int* __restrict__ src,
                                    const int* __restrict__ dst,
                                    const float* __restrict__ deginv,
                                    float* __restrict__ agg,
                                    long long total, int E) {
  long long tid = (long long)blockIdx.x * blockDim.x + threadIdx.x;
  if (tid >= total) return;
  long long edge = tid >> 6;
  int f = (int)(tid & 63);
  int s, d; float w;
  if (edge < E) {
    s = src[edge]; d = dst[edge]; w = deginv[d];
  } else {
    s = d = (int)(edge - E); w = deginv[d] * (1.0f + LAM);
  }
  atomicAdd(agg + (long long)d * 64 + f, w * x[(long long)s * 64 + f]);
}

// ---------------- GEMM: H = relu(A@Wout + b + X@Wroot), 64->64 ----------------
// grid = N/16 blocks, 128 threads (4 waves, one 16x16 tile each)
__global__ void gemm1_kernel(const float* __restrict__ A, const float* __restrict__ X,
                             const float* __restrict__ Wout,
                             const float* __restrict__ Wroot,
                             const float* __restrict__ bias,
                             float* __restrict__ H) {
  int lane = threadIdx.x & 31;
  int wave = threadIdx.x >> 5;
  int m0 = blockIdx.x * 16;
  int n0 = wave * 16;
  int r  = lane & 15;
  int kb = (lane >> 4) * 2;
  const float* Arow = A + (long long)(m0 + r) * 64;
  const float* Xrow = X + (long long)(m0 + r) * 64;
  v8f c = {};
#pragma unroll
  for (int k0 = 0; k0 < 64; k0 += 4) {
    v2f a; a.x = Arow[k0 + kb];           a.y = Arow[k0 + kb + 1];
    v2f b; b.x = Wout[(k0 + kb) * 64 + n0 + r]; b.y = Wout[(k0 + kb + 1) * 64 + n0 + r];
    c = wmma_f32_k4(a, b, c);
  }
#pragma unroll
  for (int k0 = 0; k0 < 64; k0 += 4) {
    v2f a; a.x = Xrow[k0 + kb];           a.y = Xrow[k0 + kb + 1];
    v2f b; b.x = Wroot[(k0 + kb) * 64 + n0 + r]; b.y = Wroot[(k0 + kb + 1) * 64 + n0 + r];
    c = wmma_f32_k4(a, b, c);
  }
  int hi = (lane >> 4) * 8;
  float bv = bias[n0 + r];
#pragma unroll
  for (int i = 0; i < 8; ++i) {
    float v = c[i] + bv;
    H[(long long)(m0 + i + hi) * 64 + n0 + r] = v > 0.0f ? v : 0.0f;
  }
}

// ---------------- GEMM: H = A@W, 64->64 (no bias/act) ----------------
__global__ void gemm64_kernel(const float* __restrict__ A, const float* __restrict__ W,
                              float* __restrict__ H) {
  int lane = threadIdx.x & 31;
  int wave = threadIdx.x >> 5;
  int m0 = blockIdx.x * 16;
  int n0 = wave * 16;
  int r  = lane & 15;
  int kb = (lane >> 4) * 2;
  const float* Arow = A + (long long)(m0 + r) * 64;
  v8f c = {};
#pragma unroll
  for (int k0 = 0; k0 < 64; k0 += 4) {
    v2f a; a.x = Arow[k0 + kb];        a.y = Arow[k0 + kb + 1];
    v2f b; b.x = W[(k0 + kb) * 64 + n0 + r]; b.y = W[(k0 + kb + 1) * 64 + n0 + r];
    c = wmma_f32_k4(a, b, c);
  }
  int hi = (lane >> 4) * 8;
#pragma unroll
  for (int i = 0; i < 8; ++i)
    H[(long long)(m0 + i + hi) * 64 + n0 + r] = c[i];
}

// ---------------- GEMM: H = A@W, 64->16 ----------------
// 256 threads = 8 waves, each wave one 16-row tile (uniform wave-level guard)
__global__ void gemm3_kernel(const float* __restrict__ A, const float* __restrict__ W,
                             float* __restrict__ H, int nTiles) {
  int lane = threadIdx.x & 31;
  int wave = threadIdx.x >> 5;
  int tile = blockIdx.x * 8 + wave;
  if (tile >= nTiles) return;  // wave-uniform: EXEC all-ones inside
  int m0 = tile * 16;
  int r  = lane & 15;
  int kb = (lane >> 4) * 2;
  const float* Arow = A + (long long)m0 * 64 + (long long)r * 64;
  v8f c = {};
#pragma unroll
  for (int k0 = 0; k0 < 64; k0 += 4) {
    v2f a; a.x = Arow[k0 + kb];     a.y = Arow[k0 + kb + 1];
    v2f b; b.x = W[(k0 + kb) * 16 + r]; b.y = W[(k0 + kb + 1) * 16 + r];
    c = wmma_f32_k4(a, b, c);
  }
  int hi = (lane >> 4) * 8;
#pragma unroll
  for (int i = 0; i < 8; ++i)
    H[(long long)(m0 + i + hi) * 16 + r] = c[i];
}

// ---------------- per-node attention logits ----------------
__global__ void alpha_kernel(const float* __restrict__ H, const float* __restrict__ asrc,
                             const float* __restrict__ adst,
                             float* __restrict__ As, float* __restrict__ Ad,
                             int n, int C) {
  int i = blockIdx.x * blockDim.x + threadIdx.x;
  if (i >= n) return;
  const float* row = H + (long long)i * C;
  float s = 0.0f, d = 0.0f;
  for (int f = 0; f < C; ++f) {
    float h = row[f];
    s += h * asrc[f];
    d += h * adst[f];
  }
  As[i] = s;
  Ad[i] = d;
}

// ---------------- edge softmax pass 1: segment max ----------------
__global__ void edge_max_kernel(const int* __restrict__ src, const int* __restrict__ dst,
                                const float* __restrict__ As, const float* __restrict__ Ad,
                                unsigned* __restrict__ mOrd, int E, int EN) {
  int t = blockIdx.x * blockDim.x + threadIdx.x;
  if (t >= EN) return;
  int s = (t < E) ? src[t] : t - E;
  int d = (t < E) ? dst[t] : t - E;
  float e = As[s] + Ad[d];
  e = e > 0.0f ? e : NEG_SLOPE * e;
  atomicMax(mOrd + d, f2ord(e));
}

// ---------------- edge softmax pass 2: exp + segment sum ----------------
__global__ void edge_expsum_kernel(const int* __restrict__ src, const int* __restrict__ dst,
                                   const float* __restrict__ As, const float* __restrict__ Ad,
                                   const unsigned* __restrict__ mOrd,
                                   float* __restrict__ ee, float* __restrict__ denom,
                                   int E, int EN) {
  int t = blockIdx.x * blockDim.x + threadIdx.x;
  if (t >= EN) return;
  int s = (t < E) ? src[t] : t - E;
  int d = (t < E) ? dst[t] : t - E;
  float e = As[s] + Ad[d];
  e = e > 0.0f ? e : NEG_SLOPE * e;
  float v = __expf(e - ord2f(mOrd[d]));
  ee[t] = v;
  atomicAdd(denom + d, v);
}

// ---------------- edge softmax pass 3: weighted scatter ----------------
template <int C>
__global__ void edge_scatter_kernel(const int* __restrict__ src, const int* __restrict__ dst,
                                    const float* __restrict__ H,
                                    const float* __restrict__ ee,
                                    const float* __restrict__ denom,
                                    float* __restrict__ out, int E, long long total) {
  long long tid = (long long)blockIdx.x * blockDim.x + threadIdx.x;
  if (tid >= total) return;
  long long edge = tid / C;
  int f = (int)(tid % C);
  int s = (edge < E) ? src[edge] : (int)(edge - E);
  int d = (edge < E) ? dst[edge] : (int)(edge - E);
  float a = ee[edge] / denom[d];
  atomicAdd(out + (long long)d * C + f, a * H[(long long)s * C + f]);
}

// ---------------- bias (+optional relu) ----------------
__global__ void bias_act_kernel(float* __restrict__ p, const float* __restrict__ b,
                                long long n, int C, int do_relu) {
  long long i = (long long)blockIdx.x * blockDim.x + threadIdx.x;
  if (i >= n) return;
  float v = p[i] + b[i % C];
  p[i] = (do_relu && v < 0.0f) ? 0.0f : v;
}

static inline int cdiv_ll(long long a, long long b) { return (int)((a + b - 1) / b); }

extern "C" void kernel_launch(void* const* d_in, const int* in_sizes, int n_in,
                              void* d_out, int out_size, void* d_ws, size_t ws_size,
                              hipStream_t stream) {
  const float* x   = (const float*)d_in[0];
  const int*   ei  = (const int*)d_in[1];
  const float* W1o = (const float*)d_in[2];
  const float* b1  = (const float*)d_in[3];
  const float* W1r = (const float*)d_in[4];
  const float* W2  = (const float*)d_in[5];
  const float* a2s = (const float*)d_in[6];
  const float* a2d = (const float*)d_in[7];
  const float* b2  = (const float*)d_in[8];
  const float* W3  = (const float*)d_in[9];
  const float* a3s = (const float*)d_in[10];
  const float* a3d = (const float*)d_in[11];
  const float* b3  = (const float*)d_in[12];

  const int N  = in_sizes[0] / 64;
  const int E  = in_sizes[1] / 2;
  const int EN = E + N;
  const int* src = ei;
  const int* dst = ei + E;

  // workspace layout (floats)
  float*    ws    = (float*)d_ws;
  float*    deg   = ws;                       // N   (becomes deg_inv)
  float*    As    = deg + N;                  // N
  float*    Ad    = As + N;                   // N
  unsigned* mOrd  = (unsigned*)(Ad + N);      // N
  float*    denom = (float*)(mOrd + N);       // N
  float*    ee    = denom + N;                // EN
  float*    bufA  = ee + EN;                  // N*64
  float*    bufB  = bufA + (size_t)N * 64;    // N*64
  float*    h3    = bufB + (size_t)N * 64;    // N*16
  float*    out   = (float*)d_out;            // N*16

  const int TB = 256;
  const long long N64 = (long long)N * 64;
  const long long N16 = (long long)N * 16;
  const long long EN64 = (long long)EN * 64;
  const long long EN16 = (long long)EN * 16;
  const int nTiles = N / 16;

  // ---- Layer 1: ClusterGCN ----
  fill_f32_kernel<<<cdiv_ll(N, TB), TB, 0, stream>>>(deg, N, 1.0f);
  deg_count_kernel<<<cdiv_ll(E, TB), TB, 0, stream>>>(dst, deg, E);
  recip_kernel<<<cdiv_ll(N, TB), TB, 0, stream>>>(deg, N);
  fill_f32_kernel<<<cdiv_ll(N64, TB), TB, 0, stream>>>(bufA, N64, 0.0f);
  cgcn_scatter_kernel<<<cdiv_ll(EN64, TB), TB, 0, stream>>>(x, src, dst, deg, bufA,
                                                            EN64, E);
  // h1 = relu(agg@W1_out + b1 + x@W1_root)  -> bufB
  gemm1_kernel<<<nTiles, 128, 0, stream>>>(bufA, x, W1o, W1r, b1, bufB);

  // ---- Layer 2: GAT 64->64 ----
  gemm64_kernel<<<nTiles, 128, 0, stream>>>(bufB, W2, bufA);  // h2lin -> bufA
  alpha_kernel<<<cdiv_ll(N, TB), TB, 0, stream>>>(bufA, a2s, a2d, As, Ad, N, 64);
  fill_u32_kernel<<<cdiv_ll(N, TB), TB, 0, stream>>>(mOrd, N, ORD_NEG_INF);
  fill_f32_kernel<<<cdiv_ll(N, TB), TB, 0, stream>>>(denom, N, 0.0f);
  edge_max_kernel<<<cdiv_ll(EN, TB), TB, 0, stream>>>(src, dst, As, Ad, mOrd, E, EN);
  edge_expsum_kernel<<<cdiv_ll(EN, TB), TB, 0, stream>>>(src, dst, As, Ad, mOrd, ee,
                                                         denom, E, EN);
  fill_f32_kernel<<<cdiv_ll(N64, TB), TB, 0, stream>>>(bufB, N64, 0.0f);
  edge_scatter_kernel<64><<<cdiv_ll(EN64, TB), TB, 0, stream>>>(src, dst, bufA, ee,
                                                                denom, bufB, E, EN64);
  bias_act_kernel<<<cdiv_ll(N64, TB), TB, 0, stream>>>(bufB, b2, N64, 64, 1);  // h2

  // ---- Layer 3: GAT 64->16 ----
  gemm3_kernel<<<cdiv_ll(nTiles, 8), 256, 0, stream>>>(bufB, W3, h3, nTiles);
  alpha_kernel<<<cdiv_ll(N, TB), TB, 0, stream>>>(h3, a3s, a3d, As, Ad, N, 16);
  fill_u32_kernel<<<cdiv_ll(N, TB), TB, 0, stream>>>(mOrd, N, ORD_NEG_INF);
  fill_f32_kernel<<<cdiv_ll(N, TB), TB, 0, stream>>>(denom, N, 0.0f);
  edge_max_kernel<<<cdiv_ll(EN, TB), TB, 0, stream>>>(src, dst, As, Ad, mOrd, E, EN);
  edge_expsum_kernel<<<cdiv_ll(EN, TB), TB, 0, stream>>>(src, dst, As, Ad, mOrd, ee,
                                                         denom, E, EN);
  fill_f32_kernel<<<cdiv_ll(N16, TB), TB, 0, stream>>>(out, N16, 0.0f);
  edge_scatter_kernel<16><<<cdiv_ll(EN16, TB), TB, 0, stream>>>(src, dst, h3, ee,
                                                                denom, out, E, EN16);
  bias_act_kernel<<<cdiv_ll(N16, TB), TB, 0, stream>>>(out, b3, N16, 16, 0);
}